// custom_loss_68719476907
// MI455X (gfx1250) — compile-verified
//
#include <hip/hip_runtime.h>
#include <math.h>

// ---------------------------------------------------------------------------
// YOLO-v1 loss reduction for MI455X (gfx1250, wave32).
// Pass 1: 6272 blocks x 256 threads; stage 256 cells (output+target) to LDS
//         via async global->LDS b128 (ASYNCcnt path), compute per-cell loss
//         terms, reduce 256 partials per block with V_WMMA_F32_16X16X4_F32
//         (B = ones => row-sum reduction, f32 exact), write 3 block partials
//         to d_ws.
// Pass 2: 1 block deterministically reduces the 6272x3 partials the same way.
// No float atomics => deterministic across replays.
// ---------------------------------------------------------------------------

static constexpr float K_COORD = 5.0f;
static constexpr float K_NOOBJ = 0.5f;
static constexpr float K_SINV  = 1.0f / 7.0f;

static constexpr int CPB    = 256;         // cells per block == threads per block
static constexpr int FPC    = 30;          // floats per cell
static constexpr int CHUNK  = CPB * FPC;   // 7680 floats per tensor per block
static constexpr int CHUNK4 = CHUNK / 4;   // 1920 float4 per tensor per block

typedef __attribute__((ext_vector_type(2))) float v2f;
typedef __attribute__((ext_vector_type(8))) float v8f;

// Type expected by the async-LDS builtin (from the round-1 diagnostic):
// first param is 'int __attribute__((vector_size(16))) __device__ *'.
typedef int v4i __attribute__((vector_size(16)));
typedef __attribute__((address_space(1))) v4i g_v4i;   // global (AS1)
typedef __attribute__((address_space(3))) v4i l_v4i;   // LDS (AS3)

#ifdef __has_builtin
#if __has_builtin(__builtin_amdgcn_wmma_f32_16x16x4_f32)
#define HAVE_WMMA_F32X4 1
#endif
#if __has_builtin(__builtin_amdgcn_global_load_async_to_lds_b128)
#define HAVE_ASYNC_B128 1
#endif
#if __has_builtin(__builtin_amdgcn_s_wait_asynccnt)
#define HAVE_WAIT_ASYNC 1
#endif
#endif

// Sum 256 floats living in LDS, executed by the 32 lanes of wave 0.
// WMMA path: A holds 64 values/chunk (2 VGPRs), B = ones(4x16), C chains the
// accumulation. D[m][n] = rowsum(m)+C independent of n, so summing the 8 D
// VGPRs and adding the opposite half-wave (xor 16) gives the total in every
// lane — robust to A-striping details.
__device__ __forceinline__ float wave_sum256(const float* red, int lane) {
#ifdef HAVE_WMMA_F32X4
  v8f acc = {};
  v2f ones;
  ones.x = 1.0f;
  ones.y = 1.0f;
#pragma unroll
  for (int c = 0; c < 4; ++c) {
    v2f a;
    a.x = red[c * 64 + lane];        // bank-conflict-free: stride-1 per lane
    a.y = red[c * 64 + 32 + lane];
    acc = __builtin_amdgcn_wmma_f32_16x16x4_f32(
        /*neg_a=*/false, a, /*neg_b=*/false, ones,
        /*c_mod=*/(short)0, acc, /*reuse_a=*/false, /*reuse_b=*/false);
  }
  float s = acc[0] + acc[1] + acc[2] + acc[3] + acc[4] + acc[5] + acc[6] + acc[7];
  s += __shfl_xor(s, 16, 32);
  return s;
#else
  float s = 0.0f;
#pragma unroll
  for (int c = 0; c < 8; ++c) s += red[c * 32 + lane];
#pragma unroll
  for (int m = 16; m >= 1; m >>= 1) s += __shfl_xor(s, m, 32);
  return s;
#endif
}

// Stage CHUNK4 float4s global -> LDS, coalesced b128.
__device__ __forceinline__ void stage_chunk(const float4* __restrict__ g,
                                            float4* lds, int tid) {
#ifdef HAVE_ASYNC_B128
  for (int i = tid; i < CHUNK4; i += CPB) {
    // generic -> AS1/AS3 via integer punning (LDS generic low 32 bits == LDS
    // byte offset per the gfx1250 aperture mapping).
    g_v4i* gp = (g_v4i*)(unsigned long long)(const void*)(g + i);
    l_v4i* lp = (l_v4i*)(unsigned int)(unsigned long long)(const void*)(lds + i);
    __builtin_amdgcn_global_load_async_to_lds_b128(gp, lp, 0, 0);
  }
#else
  for (int i = tid; i < CHUNK4; i += CPB) lds[i] = g[i];
#endif
}

__device__ __forceinline__ void wait_stage_done() {
#ifdef HAVE_ASYNC_B128
#ifdef HAVE_WAIT_ASYNC
  __builtin_amdgcn_s_wait_asynccnt(0);
#else
  asm volatile("s_wait_asynccnt 0" ::: "memory");
#endif
#endif
}

__global__ __launch_bounds__(CPB) void yolo_loss_pass1(
    const float* __restrict__ outp, const float* __restrict__ tgtp,
    float* __restrict__ ws) {
  __shared__ float4 s_out4[CHUNK4];
  __shared__ float4 s_tgt4[CHUNK4];
  __shared__ float s_red[3 * CPB];

  const int tid = threadIdx.x;
  const unsigned long long base = (unsigned long long)blockIdx.x * CHUNK;

  stage_chunk((const float4*)(outp + base), s_out4, tid);
  stage_chunk((const float4*)(tgtp + base), s_tgt4, tid);
  wait_stage_done();
  __syncthreads();

  const float* o = (const float*)s_out4 + tid * FPC;
  const float* t = (const float*)s_tgt4 + tid * FPC;

  const float w_obj = (t[4] > 0.0f) ? 1.0f : 0.0f;
  const float w_no  = 1.0f - w_obj;

  // ---- class loss + argmax match (first-max tie-break, like jnp.argmax) ----
  float cls = 0.0f;
  int oa = 0, ta = 0;
  float om = o[10], tm = t[10];
#pragma unroll
  for (int k = 1; k < 20; ++k) {
    const float ov = o[10 + k], tv = t[10 + k];
    if (ov > om) { om = ov; oa = k; }
    if (tv > tm) { tm = tv; ta = k; }
  }
#pragma unroll
  for (int k = 0; k < 20; ++k) {
    const float d = o[10 + k] - t[10 + k];
    cls += d * d;
  }
  cls *= w_obj;
  const float acc_cell = (oa == ta) ? w_obj : 0.0f;

  // ---- IOU of both predicted boxes vs target box 0 ----
  const float txc = t[0] * K_SINV, tyc = t[1] * K_SINV;
  const float tw = t[2], th = t[3];
  const float tlx = txc - 0.5f * tw, tly = tyc - 0.5f * th;
  const float trx = txc + 0.5f * tw, try_ = tyc + 0.5f * th;
  const float area_t = tw * th;

  float iou[2];
#pragma unroll
  for (int k = 0; k < 2; ++k) {
    const float* ob = o + 5 * k;
    const float xc = ob[0] * K_SINV, yc = ob[1] * K_SINV;
    const float w = ob[2], h = ob[3];
    const float lx = xc - 0.5f * w, ly = yc - 0.5f * h;
    const float rx = xc + 0.5f * w, ry = yc + 0.5f * h;
    const float iw = fmaxf(fminf(rx, trx) - fmaxf(lx, tlx), 0.0f);
    const float ih = fmaxf(fminf(ry, try_) - fmaxf(ly, tly), 0.0f);
    const float inter = iw * ih;
    iou[k] = inter / (w * h + area_t - inter);
  }
  const int resp = (iou[1] > iou[0]) ? 1 : 0;
  const float max_iou = fmaxf(iou[0], iou[1]);

  const float* os = o + 5 * resp;
  const float* ts = t + 5 * resp;
  const float oab4 = o[5 * (1 - resp) + 4];

  const float dx = os[0] - ts[0], dy = os[1] - ts[1];
  const float dsw = sqrtf(os[2]) - sqrtf(ts[2]);
  const float dsh = sqrtf(os[3]) - sqrtf(ts[3]);
  const float coord = w_obj * (dx * dx + dy * dy + dsw * dsw + dsh * dsh);
  const float dconf = os[4] - max_iou;
  const float conf = w_obj * dconf * dconf;
  const float aband = w_obj * oab4 * oab4;
  const float d4 = o[4] - t[4], d9 = o[9] - t[9];
  const float noobj = w_no * (d4 * d4 + d9 * d9);

  const float total = K_COORD * coord + conf + K_NOOBJ * (noobj + aband) + cls;

  s_red[tid]           = total;
  s_red[CPB + tid]     = w_obj * max_iou;
  s_red[2 * CPB + tid] = acc_cell;
  __syncthreads();

  if (tid < 32) {  // wave 0, EXEC all-ones: WMMA-legal
    const float r0 = wave_sum256(&s_red[0], tid);
    const float r1 = wave_sum256(&s_red[CPB], tid);
    const float r2 = wave_sum256(&s_red[2 * CPB], tid);
    if (tid == 0) {
      ws[blockIdx.x * 3 + 0] = r0;
      ws[blockIdx.x * 3 + 1] = r1;
      ws[blockIdx.x * 3 + 2] = r2;
    }
  }
}

__global__ __launch_bounds__(CPB) void yolo_loss_pass2(
    const float* __restrict__ ws, float* __restrict__ outv, int nblocks) {
  __shared__ float s_red[3 * CPB];
  const int tid = threadIdx.x;
  float a = 0.0f, b = 0.0f, c = 0.0f;
  for (int i = tid; i < nblocks; i += CPB) {
    a += ws[i * 3 + 0];
    b += ws[i * 3 + 1];
    c += ws[i * 3 + 2];
  }
  s_red[tid]           = a;
  s_red[CPB + tid]     = b;
  s_red[2 * CPB + tid] = c;
  __syncthreads();
  if (tid < 32) {
    const float r0 = wave_sum256(&s_red[0], tid);
    const float r1 = wave_sum256(&s_red[CPB], tid);
    const float r2 = wave_sum256(&s_red[2 * CPB], tid);
    if (tid == 0) {
      outv[0] = r0;
      outv[1] = r1;
      outv[2] = r2;
    }
  }
}

extern "C" void kernel_launch(void* const* d_in, const int* in_sizes, int n_in,
                              void* d_out, int out_size, void* d_ws, size_t ws_size,
                              hipStream_t stream) {
  (void)n_in; (void)out_size; (void)ws_size;
  const float* outp = (const float*)d_in[0];
  const float* tgtp = (const float*)d_in[1];
  float* ws = (float*)d_ws;

  const int total_cells = in_sizes[0] / FPC;       // B*7*7 = 1,605,632
  const int nblocks = total_cells / CPB;           // 6272 (exact for B=32768)

  yolo_loss_pass1<<<nblocks, CPB, 0, stream>>>(outp, tgtp, ws);
  yolo_loss_pass2<<<1, CPB, 0, stream>>>(ws, (float*)d_out, nblocks);
}